// Lca_78125455114251
// MI455X (gfx1250) — compile-verified
//
#include <hip/hip_runtime.h>

typedef float v2f __attribute__((ext_vector_type(2)));
typedef float v8f __attribute__((ext_vector_type(8)));

namespace {
constexpr int   B      = 16;
constexpr int   C      = 128;
constexpr int   L      = 16000;
constexpr int   K      = 256;
constexpr int   STRIDE = 4;
constexpr int   ITERS  = 10;
constexpr int   S      = (L - K) / STRIDE + 1;           // 3937
constexpr int   N      = B * C * S;                       // 8,062,976
constexpr int   M      = L / STRIDE;                      // 4000 m-positions per phase
constexpr float STEP   = 0.1f;                            // DT/TAU
constexpr float THR    = 0.5f;
constexpr int   S_TILE   = 32;                            // 2 WMMA tiles per wave
constexpr int   S_TILES  = (S + S_TILE - 1) / S_TILE;     // 124
constexpr int   M_CHUNK  = 32;
constexpr int   M_CHUNKS = M / M_CHUNK;                   // 125
}

__device__ __forceinline__ float hshrink(float v) {
  return (fabsf(v) > THR) ? v : 0.0f;
}

// Strided conv1d via V_WMMA_F32_16X16X4_F32.
//   out[b,c,s] = sum_k W[c,k] * xin[b, 4*s + k]
// One wave computes a 16(c) x 32(s) tile: one A-fragment (W) feeds two WMMAs
// per k-step (64 steps over K=256). The B-tile for step k0 is a contiguous
// 64-float window of the signal, staged in LDS.
// UPDATE:  fuse u += STEP*(drive - u - feedback + hardshrink(u))
// WRITE_A: additionally emit a_out = hardshrink(u_new)  (final iteration)
template <bool UPDATE, bool WRITE_A>
__global__ __launch_bounds__(256) void conv_wmma_kernel(
    const float* __restrict__ xin,    // [B, L]  (mini_batch or recon)
    const float* __restrict__ Wg,     // [C, K]
    float* __restrict__ outp,         // !UPDATE: drive [B,C,S]
    const float* __restrict__ drive,  // UPDATE
    float* __restrict__ u,            // UPDATE (in/out)
    float* __restrict__ aout)         // WRITE_A: final sparse code [B,C,S]
{
  __shared__ __align__(16) float xwin[384];  // signal window for this s-tile

  const int b    = blockIdx.y;
  const int s0   = blockIdx.x * S_TILE;
  const int tid  = threadIdx.x;
  const int wave = tid >> 5;
  const int lane = tid & 31;
  const int hi   = lane >> 4;   // half-wave select (ISA fragment layouts)
  const int lo   = lane & 15;
  const int c0   = wave * 16;   // 8 waves cover all 128 channels

  // Stage xwin[t] = xin[b, 4*s0 + t]; needed span = 4*31 + 252 + 4 = 380.
  const int base = 4 * s0;
  for (int t = tid; t < 384; t += 256) {
    const int g = base + t;
    xwin[t] = (g < L) ? xin[(size_t)b * L + g] : 0.0f;
  }

  // Prefetch epilogue cachelines (u/drive) while the WMMA loop runs.
  if (UPDATE) {
    const size_t pidx = ((size_t)b * C + c0) * S + s0 + lo;
    __builtin_prefetch(drive + pidx, 0, 1);
    __builtin_prefetch(u + pidx, 1, 1);
  }
  __syncthreads();

  // A-fragment (16x4 f32): lane lo = row M, VGPRs hold K = 2*hi, 2*hi+1.
  const float* wrow = Wg + (size_t)(c0 + lo) * K + 2 * hi;

  v8f acc[2] = {{}, {}};
#pragma unroll 4
  for (int k0 = 0; k0 < K; k0 += 4) {
    v2f afrag;
    afrag.x = wrow[k0];
    afrag.y = wrow[k0 + 1];
    // B-fragment (4x16 f32): column N = lo, rows K = 2*hi, 2*hi+1.
    // B[kk, n] = xwin[4*n + k0 + kk]
    const int bi = 4 * lo + k0 + 2 * hi;
    v2f bf0, bf1;
    bf0.x = xwin[bi];
    bf0.y = xwin[bi + 1];
    bf1.x = xwin[bi + 64];
    bf1.y = xwin[bi + 65];
    acc[0] = __builtin_amdgcn_wmma_f32_16x16x4_f32(
        false, afrag, false, bf0, (short)0, acc[0], false, false);
    acc[1] = __builtin_amdgcn_wmma_f32_16x16x4_f32(
        false, afrag, false, bf1, (short)0, acc[1], false, false);
  }

  // C/D layout: VGPR v -> row v (lanes 0-15) / v+8 (lanes 16-31), col = lo.
#pragma unroll
  for (int t2 = 0; t2 < 2; ++t2) {
    const int s = s0 + 16 * t2 + lo;
    if (s < S) {
#pragma unroll
      for (int v = 0; v < 8; ++v) {
        const int c = c0 + v + 8 * hi;
        const size_t idx = ((size_t)b * C + c) * S + s;
        const float fb = acc[t2][v];
        if (UPDATE) {
          const float uo = u[idx];
          const float av = hshrink(uo);
          const float un = uo + STEP * (drive[idx] - uo - fb + av);
          u[idx] = un;
          if (WRITE_A) aout[idx] = hshrink(un);
        } else {
          outp[idx] = fb;
        }
      }
    }
  }
}

// conv_transpose1d, phase-decomposed:
//   recon[b, 4m + r] = sum_c sum_{j=0}^{63} hardshrink(u[b,c,m-j]) * W[c, 4j + r]
// Block = 128 threads = 4 waves, one wave per phase r (weight index is
// wave-uniform -> conflict-free LDS broadcast). Handles 32 consecutive m.
// LDS: sha (48 KB hardshrunk-u window) + wtile (16 KB, one 16-channel W tile).
__global__ __launch_bounds__(128) void recon_kernel(
    const float* __restrict__ u,   // [B, C, S]
    const float* __restrict__ Wg,  // [C, K]
    float* __restrict__ recon)     // [B, L]
{
  __shared__ float sha[C][96];      // 48 KB
  __shared__ float wtile[16 * K];   // 16 KB

  const int b   = blockIdx.y;
  const int m0  = blockIdx.x * M_CHUNK;
  const int tid = threadIdx.x;

  // Stage sha[c][t] = hardshrink(u[b, c, m0 - 63 + t]), t in [0,96), zero-pad.
  for (int i = tid; i < C * 96; i += 128) {
    const int c  = i / 96;
    const int t  = i - c * 96;
    const int sp = m0 - 63 + t;
    float v = 0.0f;
    if (sp >= 0 && sp < S) v = hshrink(u[((size_t)b * C + c) * S + sp]);
    sha[c][t] = v;
  }

  const int r  = tid >> 5;  // phase: one wave per phase
  const int mi = tid & 31;

  float acc = 0.0f;
  for (int ct = 0; ct < C / 16; ++ct) {
    __syncthreads();  // also covers sha staging on first pass
    for (int i = tid; i < 16 * K; i += 128) {
      wtile[i] = Wg[(size_t)ct * 16 * K + i];
    }
    __syncthreads();
#pragma unroll 2
    for (int ci = 0; ci < 16; ++ci) {
      const int c = ct * 16 + ci;
#pragma unroll 8
      for (int j = 0; j < 64; ++j) {
        acc = fmaf(wtile[ci * K + 4 * j + r], sha[c][mi + 63 - j], acc);
      }
    }
  }

  const int m = m0 + mi;
  if (m < M) recon[(size_t)b * L + 4 * m + r] = acc;
}

__global__ void zero_kernel(float* __restrict__ p, int n) {
  const int i = blockIdx.x * blockDim.x + threadIdx.x;
  if (i < n) p[i] = 0.0f;
}

extern "C" void kernel_launch(void* const* d_in, const int* in_sizes, int n_in,
                              void* d_out, int out_size, void* d_ws, size_t ws_size,
                              hipStream_t stream) {
  const float* x  = (const float*)d_in[0];  // mini_batch [16,1,16000]
  const float* W  = (const float*)d_in[1];  // weights    [128,1,256]
  float* out = (float*)d_out;               // a          [16,128,3937]

  // Workspace: drive (N) | u (N) | recon (B*L). ~65.5 MB total, L2-resident.
  float* ws    = (float*)d_ws;
  float* drive = ws;
  float* u     = ws + (size_t)N;
  float* recon = ws + 2 * (size_t)N;

  const dim3 cgrid(S_TILES, B);
  const dim3 rgrid(M_CHUNKS, B);

  zero_kernel<<<(N + 255) / 256, 256, 0, stream>>>(u, N);
  conv_wmma_kernel<false, false><<<cgrid, 256, 0, stream>>>(
      x, W, drive, nullptr, nullptr, nullptr);

  for (int it = 0; it < ITERS; ++it) {
    recon_kernel<<<rgrid, 128, 0, stream>>>(u, W, recon);
    if (it < ITERS - 1) {
      conv_wmma_kernel<true, false><<<cgrid, 256, 0, stream>>>(
          recon, W, nullptr, drive, u, nullptr);
    } else {
      // Final iteration: fuse a = hardshrink(u_new) into the update epilogue.
      conv_wmma_kernel<true, true><<<cgrid, 256, 0, stream>>>(
          recon, W, nullptr, drive, u, out);
    }
  }
}